// SpatialMTP1Hop_46420006535686
// MI455X (gfx1250) — compile-verified
//
#include <hip/hip_runtime.h>

typedef __attribute__((ext_vector_type(2))) float v2f;
typedef __attribute__((ext_vector_type(8))) float v8f;

// ---------------------------------------------------------------------------
// Kernel 1: init workspace. owner_of <- sentinel C, loss/cnt <- 0, out <- 0.
// ---------------------------------------------------------------------------
__global__ void init_ws(int* __restrict__ owner_of, float* __restrict__ loss_sum,
                        float* __restrict__ cnt, float* __restrict__ out, int N, int C) {
    int i = blockIdx.x * blockDim.x + threadIdx.x;
    if (i < N) owner_of[i] = C;
    if (i < C) { loss_sum[i] = 0.0f; cnt[i] = 0.0f; }
    if (i == 0) out[0] = 0.0f;
}

// ---------------------------------------------------------------------------
// Kernel 2: owner_of[centers[i]] = i
// ---------------------------------------------------------------------------
__global__ void scatter_owner(const int* __restrict__ centers, int* __restrict__ owner_of,
                              int C) {
    int i = blockIdx.x * blockDim.x + threadIdx.x;
    if (i < C) owner_of[centers[i]] = i;
}

// ---------------------------------------------------------------------------
// Kernel 3 (D known at compile time): node_err[n] =
//     mean_j ( (H[n,:] @ W[:,j] + b[j] - target[n,j])^2 )
// One wave32 per 16-row tile; fp32 WMMA 16x16x4 accumulation over K = D.
//
// A (16x4 f32) VGPR layout: lanes 0-15 hold {K=0,K=1}, lanes 16-31 hold
// {K=2,K=3}, M = lane&15. B (4x16) mirrors it with N = lane&15. C/D (16x16
// f32): VGPR r, lanes 0-15 -> (M=r, N=lane), lanes 16-31 -> (M=r+8, N=lane-16).
//
// The zero-padded B panel (W with DO valid columns out of 16) is staged in
// LDS, pair-major: Blds[p*32 + 2n + j] = Wpad[2p + j][n]  (p = k/2, j = k&1).
// Each lane's B operand per k-step is then one contiguous 8-byte ds_load.
// ---------------------------------------------------------------------------
template<int D>
__launch_bounds__(256)
__global__ void node_err_wmma(const float* __restrict__ H, const float* __restrict__ W,
                              const float* __restrict__ bvec, const float* __restrict__ target,
                              float* __restrict__ node_err, int N, int DO) {
    __shared__ float Blds[D * 16];              // D=128 -> 8 KB

    // Cooperative fill of the zero-padded, pair-major B panel.
    for (int idx = threadIdx.x; idx < D * 16; idx += blockDim.x) {
        const int p = idx >> 5;                 // k-pair
        const int r = idx & 31;
        const int n = r >> 1;                   // B column 0..15
        const int j = r & 1;
        const int k = 2 * p + j;
        Blds[idx] = (n < DO) ? W[k * DO + n] : 0.0f;
    }
    __syncthreads();

    const int lane = threadIdx.x & 31;
    const int wave = threadIdx.x >> 5;
    const int tile = blockIdx.x * (blockDim.x >> 5) + wave;
    if (tile * 16 >= N) return;                 // uniform per wave: EXEC stays all-1s

    const int m  = lane & 15;                   // A row within tile / B column
    const int kh = lane >> 4;                   // which K-half of the 16x4 step

    int rowA = tile * 16 + m;
    if (rowA >= N) rowA = N - 1;                // clamp; tail error is zeroed below
    const float* __restrict__ Hrow = H + (size_t)rowA * (size_t)D;

    const float bn = (m < DO) ? bvec[m] : 0.0f;

    v8f acc = {};
#pragma unroll
    for (int k4 = 0; k4 < D; k4 += 4) {
        const int k0 = k4 + 2 * kh;             // even
        const v2f a = *(const v2f*)(Hrow + k0);                 // global_load_b64
        const v2f b = *(const v2f*)(Blds + (k0 >> 1) * 32 + 2 * m); // ds_load_b64
        // D = A(16x4) * B(4x16) + C, full fp32
        acc = __builtin_amdgcn_wmma_f32_16x16x4_f32(false, a, false, b,
                                                    (short)0, acc, false, false);
    }

#pragma unroll
    for (int r = 0; r < 8; ++r) {
        const int M   = r + 8 * kh;
        const int row = tile * 16 + M;
        float e = 0.0f;
        if (m < DO && row < N) {
            const float diff = (acc[r] + bn) - target[(size_t)row * DO + m];
            e = diff * diff;
        }
        // sum the DO valid columns across the 16-lane half (lanes m>=DO hold 0)
        e += __shfl_xor(e, 1, 32);
        e += __shfl_xor(e, 2, 32);
        e += __shfl_xor(e, 4, 32);
        e += __shfl_xor(e, 8, 32);
        if (m == 0 && row < N) node_err[row] = e / (float)DO;
    }
}

// ---------------------------------------------------------------------------
// Kernel 3b: runtime-D scalar fallback (only used if D != 128).
// ---------------------------------------------------------------------------
__global__ void node_err_scalar(const float* __restrict__ H, const float* __restrict__ W,
                                const float* __restrict__ bvec, const float* __restrict__ target,
                                float* __restrict__ node_err, int N, int D, int DO) {
    int n = blockIdx.x * blockDim.x + threadIdx.x;
    if (n >= N) return;
    const float* __restrict__ Hrow = H + (size_t)n * D;
    float e = 0.0f;
    for (int j = 0; j < DO; ++j) {
        float s = bvec[j];
        for (int k = 0; k < D; ++k) s += Hrow[k] * W[k * DO + j];
        const float diff = s - target[(size_t)n * DO + j];
        e += diff * diff;
    }
    node_err[n] = e / (float)DO;
}

// ---------------------------------------------------------------------------
// Kernel 4: per-edge scatter. owner = owner_of[src]; if valid, accumulate
// node_err[dst] into loss_sum[owner] and bump cnt[owner].
// owner_of / node_err / loss_sum / cnt are all << 192MB L2: gathers stay on-chip.
// ---------------------------------------------------------------------------
__global__ void edge_accum(const int* __restrict__ edge_index, const int* __restrict__ owner_of,
                           const float* __restrict__ node_err, float* __restrict__ loss_sum,
                           float* __restrict__ cnt, int E, int C) {
    int e = blockIdx.x * blockDim.x + threadIdx.x;
    if (e >= E) return;
    const int s = edge_index[e];        // row 0: src
    const int d = edge_index[E + e];    // row 1: dst
    const int ow = owner_of[s];
    if (ow < C) {
        atomicAdd(&loss_sum[ow], node_err[d]);
        atomicAdd(&cnt[ow], 1.0f);
    }
}

// ---------------------------------------------------------------------------
// Kernel 5: mean over centers of loss_sum / max(cnt, 1).
// ---------------------------------------------------------------------------
__global__ void final_reduce(const float* __restrict__ loss_sum, const float* __restrict__ cnt,
                             float* __restrict__ out, int C) {
    __shared__ float red[256];
    float v = 0.0f;
    for (int i = blockIdx.x * blockDim.x + threadIdx.x; i < C;
         i += gridDim.x * blockDim.x) {
        v += loss_sum[i] / fmaxf(cnt[i], 1.0f);
    }
    red[threadIdx.x] = v;
    __syncthreads();
    for (int s = 128; s > 0; s >>= 1) {
        if (threadIdx.x < s) red[threadIdx.x] += red[threadIdx.x + s];
        __syncthreads();
    }
    if (threadIdx.x == 0) atomicAdd(out, red[0] / (float)C);
}

// ---------------------------------------------------------------------------
// Host-side launcher
// ---------------------------------------------------------------------------
extern "C" void kernel_launch(void* const* d_in, const int* in_sizes, int n_in,
                              void* d_out, int out_size, void* d_ws, size_t ws_size,
                              hipStream_t stream) {
    const float* H          = (const float*)d_in[0];
    const float* W          = (const float*)d_in[1];
    const float* bvec       = (const float*)d_in[2];
    const float* target     = (const float*)d_in[3];
    const int*   edge_index = (const int*)d_in[4];
    const int*   centers    = (const int*)d_in[5];

    const int DO = in_sizes[2];             // 3
    const int D  = in_sizes[1] / DO;        // 128
    const int N  = in_sizes[0] / D;         // 100000
    const int E  = in_sizes[4] / 2;         // 1600000
    const int C  = in_sizes[5];             // 50000

    // workspace layout: [owner_of: N ints][node_err: N f32][loss_sum: C f32][cnt: C f32]
    int*   owner_of = (int*)d_ws;
    float* node_err = (float*)(owner_of + N);
    float* loss_sum = node_err + N;
    float* cnt      = loss_sum + C;
    float* out      = (float*)d_out;

    const int maxNC = (N > C) ? N : C;
    init_ws<<<(maxNC + 255) / 256, 256, 0, stream>>>(owner_of, loss_sum, cnt, out, N, C);
    scatter_owner<<<(C + 255) / 256, 256, 0, stream>>>(centers, owner_of, C);

    if (D == 128) {
        const int tiles  = (N + 15) / 16;
        const int blocks = (tiles + 7) / 8;     // 8 waves per 256-thread block
        node_err_wmma<128><<<blocks, 256, 0, stream>>>(H, W, bvec, target, node_err, N, DO);
    } else {
        node_err_scalar<<<(N + 255) / 256, 256, 0, stream>>>(H, W, bvec, target,
                                                             node_err, N, D, DO);
    }

    edge_accum<<<(E + 255) / 256, 256, 0, stream>>>(edge_index, owner_of, node_err,
                                                    loss_sum, cnt, E, C);
    final_reduce<<<64, 256, 0, stream>>>(loss_sum, cnt, out, C);
}